// PraxisGraph_20100446945363
// MI455X (gfx1250) — compile-verified
//
#include <hip/hip_runtime.h>
#include <math.h>

// Problem constants (match reference)
#define B_   4
#define S_   4096
#define H_   2048
#define E_   16
#define ED_  512
#define NCHUNK 128          // S_ / RPB
#define RPB    32           // rows per block in LN pass
#define LN_EPS 1e-5f

typedef float v2f __attribute__((ext_vector_type(2)));
typedef float v8f __attribute__((ext_vector_type(8)));

// ---------------------------------------------------------------------------
// Kernel 1: fused LayerNorm + partial mean over S.
// partial[b][chunk][h] = sum_{s in chunk} (x[b,s,h]-mu)*rsqrt(var+eps)
// 4 rows per barrier phase -> 32KB/block in flight (latency hiding).
// ---------------------------------------------------------------------------
__global__ __launch_bounds__(256)
void k_ln_mean(const float* __restrict__ x, float* __restrict__ partial) {
    const int t     = threadIdx.x;
    const int b     = blockIdx.x / NCHUNK;
    const int chunk = blockIdx.x % NCHUNK;
    const int row0  = chunk * RPB;
    const int lane  = t & 31;
    const int wid   = t >> 5;

    __shared__ float2 red[4][8];

    float4 acc0 = {0.f, 0.f, 0.f, 0.f};
    float4 acc1 = {0.f, 0.f, 0.f, 0.f};

    for (int r = 0; r < RPB; r += 4) {
        float4 v0[4], v1[4];
        #pragma unroll
        for (int j = 0; j < 4; ++j) {
            const float4* p = (const float4*)(x + (size_t)(b * S_ + row0 + r + j) * H_);
            v0[j] = p[t];
            v1[j] = p[t + 256];
        }
        float s[4], ss[4];
        #pragma unroll
        for (int j = 0; j < 4; ++j) {
            s[j]  = v0[j].x + v0[j].y + v0[j].z + v0[j].w
                  + v1[j].x + v1[j].y + v1[j].z + v1[j].w;
            ss[j] = v0[j].x*v0[j].x + v0[j].y*v0[j].y + v0[j].z*v0[j].z + v0[j].w*v0[j].w
                  + v1[j].x*v1[j].x + v1[j].y*v1[j].y + v1[j].z*v1[j].z + v1[j].w*v1[j].w;
        }
        #pragma unroll
        for (int off = 16; off > 0; off >>= 1) {
            #pragma unroll
            for (int j = 0; j < 4; ++j) {       // 4 independent chains for ILP
                s[j]  += __shfl_xor(s[j],  off, 32);
                ss[j] += __shfl_xor(ss[j], off, 32);
            }
        }
        if (lane == 0) {
            #pragma unroll
            for (int j = 0; j < 4; ++j) red[j][wid] = make_float2(s[j], ss[j]);
        }
        __syncthreads();
        #pragma unroll
        for (int j = 0; j < 4; ++j) {
            float st = 0.f, sst = 0.f;
            #pragma unroll
            for (int i = 0; i < 8; ++i) { st += red[j][i].x; sst += red[j][i].y; }
            float mu  = st * (1.0f / H_);
            float var = sst * (1.0f / H_) - mu * mu;
            float rs  = rsqrtf(var + LN_EPS);
            acc0.x += (v0[j].x - mu) * rs;  acc0.y += (v0[j].y - mu) * rs;
            acc0.z += (v0[j].z - mu) * rs;  acc0.w += (v0[j].w - mu) * rs;
            acc1.x += (v1[j].x - mu) * rs;  acc1.y += (v1[j].y - mu) * rs;
            acc1.z += (v1[j].z - mu) * rs;  acc1.w += (v1[j].w - mu) * rs;
        }
        __syncthreads();
    }
    float4* out = (float4*)(partial + (size_t)(b * NCHUNK + chunk) * H_);
    out[t]       = acc0;
    out[t + 256] = acc1;
}

// ---------------------------------------------------------------------------
// Kernel 2: reduce chunk partials -> hidden_mean; build query_input (padded to
// 16 rows, rows 4..15 zero) and layer_features. Partials are L2-resident.
// ---------------------------------------------------------------------------
__global__ __launch_bounds__(256)
void k_finalize(const float* __restrict__ partial,
                const float* __restrict__ layer_emb,
                const float* __restrict__ cent_emb,
                const float* __restrict__ gamma,
                const float* __restrict__ beta,
                const int*   __restrict__ cur_p,
                float* __restrict__ qin,   // [16][H]
                float* __restrict__ lf) {  // [16][H]
    const int h   = blockIdx.x * 256 + threadIdx.x;   // grid = H_/256
    const int cur = cur_p[0];
    const float g  = gamma[h];
    const float be = beta[h];
    const float ce = layer_emb[(size_t)cur * H_ + h];

    #pragma unroll
    for (int b = 0; b < B_; ++b) {
        float s = 0.f;
        #pragma unroll 8
        for (int c = 0; c < NCHUNK; ++c)
            s += partial[(size_t)(b * NCHUNK + c) * H_ + h];
        float m = s * (1.0f / S_);
        qin[b * H_ + h] = ce + g * m + be;
    }
    #pragma unroll
    for (int b = B_; b < 16; ++b) qin[b * H_ + h] = 0.f;
    #pragma unroll
    for (int e = 0; e < E_; ++e)
        lf[e * H_ + h] = layer_emb[(size_t)e * H_ + h] + cent_emb[(size_t)e * H_ + h];
}

// ---------------------------------------------------------------------------
// Kernel 3: k-split GEMM partials via V_WMMA_F32_16X16X4_F32.
// grid = (16 n-blocks, 32 k-splits, 2 matrices) = 1024 blocks; each block
// does ONE staging phase (16x64 A + 128x64 W, float4 coalesced, pitch 68 =>
// conflict-free) then 16 WMMAs/wave; writes P[ks][16][H].
// ---------------------------------------------------------------------------
#define KS  32
#define BK2 64
#define WP  68
__global__ __launch_bounds__(256)
void k_gemm16_ks(const float* __restrict__ Aq, const float* __restrict__ Wq,
                 float* __restrict__ Pq,
                 const float* __restrict__ Ak, const float* __restrict__ Wk,
                 float* __restrict__ Pk) {
    const float* A = (blockIdx.z == 0) ? Aq : Ak;
    const float* W = (blockIdx.z == 0) ? Wq : Wk;
    float*       P = (blockIdx.z == 0) ? Pq : Pk;

    __shared__ float Asm[16 * WP];
    __shared__ float Wsm[128 * WP];

    const int t      = threadIdx.x;
    const int lane   = t & 31;
    const int wid    = t >> 5;
    const int nblock = blockIdx.x * 128;
    const int k0     = blockIdx.y * BK2;

    {   // stage A tile 16x64 (one float4 per thread)
        int r = t >> 4, c = (t & 15) * 4;
        *(float4*)&Asm[r * WP + c] = *(const float4*)&A[(size_t)r * H_ + k0 + c];
    }
    {   // stage W tile 128x64 (8 float4 per thread, 256B contiguous per half-wave)
        int r0 = t >> 4, c = (t & 15) * 4;
        #pragma unroll
        for (int i = 0; i < 8; ++i) {
            int r = r0 + i * 16;
            *(float4*)&Wsm[r * WP + c] =
                *(const float4*)&W[(size_t)(nblock + r) * H_ + k0 + c];
        }
    }
    __syncthreads();

    const int m    = lane & 15;
    const int koff = (lane < 16) ? 0 : 2;
    const float* arow = &Asm[m * WP];
    const float* wrow = &Wsm[(wid * 16 + m) * WP];

    v8f c = {};
    #pragma unroll
    for (int kk = 0; kk < BK2; kk += 4) {
        v2f a  = *(const v2f*)(arow + kk + koff);
        v2f bb = *(const v2f*)(wrow + kk + koff);
        c = __builtin_amdgcn_wmma_f32_16x16x4_f32(false, a, false, bb,
                                                  (short)0, c, false, false);
    }

    const int col   = nblock + wid * 16 + m;
    const int rbase = (lane < 16) ? 0 : 8;
    #pragma unroll
    for (int v = 0; v < 8; ++v)
        P[(size_t)(blockIdx.y * 16 + rbase + v) * H_ + col] = c[v];
}

// ---------------------------------------------------------------------------
// Kernel 3b: reduce k-split partials + bias -> q / k.  L2-resident.
// grid = (H/256, 16 rows, 2 matrices)
// ---------------------------------------------------------------------------
__global__ __launch_bounds__(256)
void k_reduce_bias(const float* __restrict__ Pq, const float* __restrict__ bq,
                   float* __restrict__ qo,
                   const float* __restrict__ Pk, const float* __restrict__ bk,
                   float* __restrict__ ko) {
    const float* P    = (blockIdx.z == 0) ? Pq : Pk;
    const float* bias = (blockIdx.z == 0) ? bq : bk;
    float*       out  = (blockIdx.z == 0) ? qo : ko;

    const int h = blockIdx.x * 256 + threadIdx.x;
    const int r = blockIdx.y;
    float s = bias[h];
    #pragma unroll 8
    for (int ks = 0; ks < KS; ++ks)
        s += P[(size_t)(ks * 16 + r) * H_ + h];
    out[(size_t)r * H_ + h] = s;
}

// ---------------------------------------------------------------------------
// Kernel 4: attention (q@k^T via WMMA, K split over 16 waves), edge Gram
// (edge@edge^T, symmetric -> A and B lane operands identical), biases, mask,
// softmax, KL loss, argmax. One block, 512 threads.
// ---------------------------------------------------------------------------
#define NW 16
__global__ __launch_bounds__(32 * NW)
void k_scores(const float* __restrict__ q,       // [16][H], rows 0..3 valid
              const float* __restrict__ km,      // [16][H]
              const float* __restrict__ edge,    // [16][ED]
              const float* __restrict__ spatial, // [E+1]
              const int*   __restrict__ cur_p,
              const int*   __restrict__ avail,
              float* __restrict__ outv) {        // [66]
    __shared__ float att_part[NW * 256];
    __shared__ float ew_part[NW * 256];

    const int t    = threadIdx.x;
    const int lane = t & 31;
    const int wid  = t >> 5;
    const int m    = lane & 15;
    const int koff = (lane < 16) ? 0 : 2;

    v8f ca = {};
    v8f ce = {};

    {   // attention partial: this wave covers H_/NW = 128 k-values
        const float* qrow = q  + (size_t)m * H_;
        const float* krow = km + (size_t)m * H_;
        const int h0b = wid * (H_ / NW);
        #pragma unroll 4
        for (int h0 = h0b; h0 < h0b + (H_ / NW); h0 += 4) {
            v2f a = *(const v2f*)(qrow + h0 + koff);
            v2f b = *(const v2f*)(krow + h0 + koff);
            ca = __builtin_amdgcn_wmma_f32_16x16x4_f32(false, a, false, b,
                                                       (short)0, ca, false, false);
        }
    }
    {   // edge Gram partial: ED_/NW = 32 k-values
        const float* erow = edge + (size_t)m * ED_;
        const int h0b = wid * (ED_ / NW);
        #pragma unroll 4
        for (int h0 = h0b; h0 < h0b + (ED_ / NW); h0 += 4) {
            v2f a = *(const v2f*)(erow + h0 + koff);
            ce = __builtin_amdgcn_wmma_f32_16x16x4_f32(false, a, false, a,
                                                       (short)0, ce, false, false);
        }
    }

    #pragma unroll
    for (int v = 0; v < 8; ++v) {
        att_part[wid * 256 + v * 32 + lane] = ca[v];
        ew_part [wid * 256 + v * 32 + lane] = ce[v];
    }
    __syncthreads();
    if (wid != 0) return;

    float catt[8], cew[8];
    #pragma unroll
    for (int v = 0; v < 8; ++v) {
        float s = 0.f, s2 = 0.f;
        for (int w = 0; w < NW; ++w) {
            s  += att_part[w * 256 + v * 32 + lane];
            s2 += ew_part [w * 256 + v * 32 + lane];
        }
        catt[v] = s;  cew[v] = s2;
    }

    const int cur = cur_p[0];

    float eb = 0.f;
    const int rbase = (lane < 16) ? 0 : 8;
    #pragma unroll
    for (int v = 0; v < 8; ++v) {
        int i = rbase + v;
        int d = (i > cur) ? (i - cur) : (cur - i);
        float coeff = (avail[i] != 0 && i != cur) ? (1.0f / (float)(d > 1 ? d : 1)) : 0.f;
        eb += coeff * cew[v];
    }
    eb += __shfl_xor(eb, 16, 32);   // combine row halves -> edge_bias[m] everywhere

    const int   dm = (m > cur) ? (m - cur) : (cur - m);
    const float sb = spatial[dm];
    const bool  av = (avail[m] != 0);
    const float rsH = rsqrtf((float)H_);

    float numav = 0.f;
    for (int e = 0; e < E_; ++e) numav += (avail[e] != 0) ? 1.f : 0.f;
    numav = fmaxf(numav, 1.f);
    const float tp   = 1.0f / numav;
    const float logt = logf(tp);

    float klacc = 0.f;
    float p0v = 0.f;
    #pragma unroll
    for (int b = 0; b < B_; ++b) {
        float sc = catt[b] * rsH + sb + eb;
        sc = av ? sc : -1e9f;
        float mx = sc;
        #pragma unroll
        for (int off = 1; off < 16; off <<= 1) mx = fmaxf(mx, __shfl_xor(mx, off, 32));
        float ex = expf(sc - mx);
        float sm = ex;
        #pragma unroll
        for (int off = 1; off < 16; off <<= 1) sm += __shfl_xor(sm, off, 32);
        float p = ex / sm;
        if (lane < 16) outv[1 + b * 16 + m] = p;
        klacc += tp * (logt - logf(fmaxf(p, 1e-10f)));
        if (b == 0) p0v = p;
    }
    #pragma unroll
    for (int off = 1; off < 16; off <<= 1) klacc += __shfl_xor(klacc, off, 32);
    if (t == 0) outv[0] = klacc * (1.0f / B_) * 0.01f;

    float bvv = p0v; int bii = m;
    #pragma unroll
    for (int off = 1; off < 16; off <<= 1) {
        float ov = __shfl_xor(bvv, off, 32);
        int   oi = __shfl_xor(bii, off, 32);
        if (ov > bvv || (ov == bvv && oi < bii)) { bvv = ov; bii = oi; }
    }
    if (t == 0) outv[65] = (float)bii;
}

// ---------------------------------------------------------------------------
extern "C" void kernel_launch(void* const* d_in, const int* in_sizes, int n_in,
                              void* d_out, int out_size, void* d_ws, size_t ws_size,
                              hipStream_t stream) {
    const float* hidden  = (const float*)d_in[0];
    const float* lemb    = (const float*)d_in[1];
    const float* cemb    = (const float*)d_in[2];
    const float* spatial = (const float*)d_in[3];
    const float* edge    = (const float*)d_in[4];
    const float* gamma   = (const float*)d_in[5];
    const float* beta    = (const float*)d_in[6];
    const float* Wq      = (const float*)d_in[7];
    const float* bq      = (const float*)d_in[8];
    const float* Wk      = (const float*)d_in[9];
    const float* bk      = (const float*)d_in[10];
    // d_in[11], d_in[12] (Wv, bv): dead in the reference -> never touched
    const int*   cur     = (const int*)d_in[13];
    const int*   avail   = (const int*)d_in[14];

    const size_t SLAB = (size_t)KS * 16 * H_;   // 1,048,576 floats (== LN partial size)

    float* ws      = (float*)d_ws;
    float* partial = ws;            // LN partials [B][NCHUNK][H] (4MB)
    float* Pq      = ws;            // aliased: reused after k_finalize consumed it
    float* Pk      = ws + SLAB;
    float* qin     = ws + 2 * SLAB;
    float* lf      = qin + 16 * H_;
    float* qo      = lf  + 16 * H_;
    float* ko      = qo  + 16 * H_;

    k_ln_mean   <<<B_ * NCHUNK, 256, 0, stream>>>(hidden, partial);
    k_finalize  <<<H_ / 256,    256, 0, stream>>>(partial, lemb, cemb, gamma, beta,
                                                  cur, qin, lf);
    k_gemm16_ks <<<dim3(H_ / 128, KS, 2), 256, 0, stream>>>(qin, Wq, Pq,
                                                            lf,  Wk, Pk);
    k_reduce_bias<<<dim3(H_ / 256, 16, 2), 256, 0, stream>>>(Pq, bq, qo,
                                                             Pk, bk, ko);
    k_scores    <<<1, 32 * NW, 0, stream>>>(qo, ko, edge, spatial, cur, avail,
                                            (float*)d_out);
}